// Loss_Function_90452011253875
// MI455X (gfx1250) — compile-verified
//
#include <hip/hip_runtime.h>
#include <math.h>

// MI455X / gfx1250, wave32. Two-pass streaming loss reduction:
//   pass A: pairwise d2 via V_WMMA_F32_16X16X4_F32 (||p||^2+||t||^2-2p.t trick),
//           packed (d2,n) u64 argmin per (b,g) (native v_min_u64) + cond mask on
//           the VALU, co-executing with the matrix pipe. No DS ops in hot loop:
//           half-waves load their A-fragment rows directly (L0 broadcast).
//   pass B: pos/gt_prob, regression + focal loss, deterministic tree reductions.

typedef float v2f __attribute__((ext_vector_type(2)));
typedef float v8f __attribute__((ext_vector_type(8)));

constexpr int   BB   = 64;
constexpr int   NN   = 16384;
constexpr int   GG   = 24;
constexpr float THT  = 3.0f / 90.0f;
constexpr float THR  = 20.0f / 400.0f;
constexpr float PADV = -1000.0f;
constexpr int   CHUNKS = 16;                  // blocks per batch image
constexpr int   ROWS_PER_BLOCK = NN / CHUNKS; // 1024 rows per block

__device__ __forceinline__ unsigned long long packMin(float d, unsigned n) {
  // d >= 0 (sum of squares): float bits are monotone as u32. Low 32 bits = n,
  // so ties resolve to the smallest n (matches jnp.argmin first-occurrence).
  return (((unsigned long long)__float_as_uint(d)) << 32) | (unsigned long long)n;
}
__device__ __forceinline__ unsigned long long minu64(unsigned long long a, unsigned long long b) {
  return a < b ? a : b;   // lowers to v_min_u64 on gfx1250
}
__device__ __forceinline__ unsigned long long shflxor64(unsigned long long v, int m) {
  unsigned lo = (unsigned)(v & 0xffffffffull);
  unsigned hi = (unsigned)(v >> 32);
  lo = __shfl_xor(lo, m, 32);
  hi = __shfl_xor(hi, m, 32);
  return (((unsigned long long)hi) << 32) | (unsigned long long)lo;
}

__global__ void lf_init(unsigned long long* minbuf, unsigned* anyPos) {
  int i = blockIdx.x * blockDim.x + threadIdx.x;
  if (i < BB * GG) minbuf[i] = ~0ull;
  if (i < BB) anyPos[i] = 0u;
}

__global__ void __launch_bounds__(256)
lf_passA(const float* __restrict__ pinit, const float* __restrict__ tgtp,
         unsigned long long* __restrict__ minbuf, unsigned* __restrict__ anyPos) {
  const int b     = blockIdx.x / CHUNKS;
  const int chunk = blockIdx.x % CHUNKS;
  __shared__ float sT[GG], sR[GG];
  const int t = threadIdx.x;
  if (t < GG) {
    float th = tgtp[(b * GG + t) * 2 + 0];
    float ra = tgtp[(b * GG + t) * 2 + 1];
    sT[t] = (th + 90.0f) * (1.0f / 180.0f);
    sR[t] = (ra + 400.0f) * (1.0f / 800.0f);
  }
  __syncthreads();

  float tT[GG], tR[GG];
#pragma unroll
  for (int g = 0; g < GG; ++g) { tT[g] = sT[g]; tR[g] = sR[g]; }

  const int  wave = t >> 5;
  const int  lane = t & 31;
  const int  j    = lane & 15;
  const bool hiH  = lane >= 16;

  // B fragments (4x16, constant over the row loop). Lanes 0-15: rows K0/K1,
  // lanes 16-31: rows K2/K3. Columns j map to g = gbase + j; pad g>=24.
  v2f bf0, bf1;
  {
    float tt0 = sT[j], rr0 = sR[j];
    bf0.x = hiH ? 1.0f : (-2.0f * tt0);
    bf0.y = hiH ? (tt0 * tt0 + rr0 * rr0) : (-2.0f * rr0);
    int   g1  = 16 + j;
    float tt1 = (g1 < GG) ? sT[g1] : 3.0f;   // pad columns -> large d2, never written
    float rr1 = (g1 < GG) ? sR[g1] : 3.0f;
    bf1.x = hiH ? 1.0f : (-2.0f * tt1);
    bf1.y = hiH ? (tt1 * tt1 + rr1 * rr1) : (-2.0f * rr1);
  }

  unsigned long long min0 = ~0ull, min1 = ~0ull;  // per-column packed (d2,n) mins
  unsigned condAcc = 0u;

  const float2* pinit2 = (const float2*)pinit;
  const int nBase = chunk * ROWS_PER_BLOCK + wave * 128;

  // Prefetch the wave's whole 128-row slab. Each half-wave loads the rows its
  // A-fragment needs (low half: rows j, high half same addresses -> L0
  // broadcast; no cross-lane DS permutes required).
  float2 av[4], cv4[4];
#pragma unroll
  for (int it = 0; it < 4; ++it) {
    const int n0 = nBase + it * 32;
    av[it]  = pinit2[(size_t)b * NN + n0 + j];        // rows n0 .. n0+15
    cv4[it] = pinit2[(size_t)b * NN + n0 + 16 + j];   // rows n0+16 .. n0+31
  }

#pragma unroll
  for (int it = 0; it < 4; ++it) {
    const int n0 = nBase + it * 32;
    const float2 a = av[it];
    const float2 c = cv4[it];

    // cond row ownership: low half covers n0+j, high half covers n0+16+j,
    // i.e. exactly row n0+lane per lane.
    const float pvx = hiH ? c.x : a.x;
    const float pvy = hiH ? c.y : a.y;
    unsigned m = 0u;
#pragma unroll
    for (int g = 0; g < GG; ++g) {
      float dt = fabsf(tT[g] - pvx);
      float dr = fabsf(tR[g] - pvy);
      m |= ((unsigned)((dt < THT) && (dr < THR))) << g;
    }
    condAcc |= m;

    // A fragments straight from the loaded rows (no shuffles).
    v2f a0, a1;
    a0.x = hiH ? (a.x * a.x + a.y * a.y) : a.x;  a0.y = hiH ? 1.0f : a.y;
    a1.x = hiH ? (c.x * c.x + c.y * c.y) : c.x;  a1.y = hiH ? 1.0f : c.y;

    v8f cz = {0.f, 0.f, 0.f, 0.f, 0.f, 0.f, 0.f, 0.f};
    v8f d00 = __builtin_amdgcn_wmma_f32_16x16x4_f32(false, a0, false, bf0, (short)0, cz, false, false);
    v8f d01 = __builtin_amdgcn_wmma_f32_16x16x4_f32(false, a0, false, bf1, (short)0, cz, false, false);
    v8f d10 = __builtin_amdgcn_wmma_f32_16x16x4_f32(false, a1, false, bf0, (short)0, cz, false, false);
    v8f d11 = __builtin_amdgcn_wmma_f32_16x16x4_f32(false, a1, false, bf1, (short)0, cz, false, false);

    // D layout: VGPR v, lanes 0-15 -> M=v, lanes 16-31 -> M=v+8, column = lane&15.
#pragma unroll
    for (int v = 0; v < 8; ++v) {
      unsigned r0 = (unsigned)(n0 + v + (hiH ? 8 : 0));
      unsigned r1 = r0 + 16u;
      min0 = minu64(min0, packMin(d00[v], r0));
      min1 = minu64(min1, packMin(d01[v], r0));
      min0 = minu64(min0, packMin(d10[v], r1));
      min1 = minu64(min1, packMin(d11[v], r1));
    }
  }

  // Combine the two row-halves of each column, OR-reduce cond mask over the wave.
  min0 = minu64(min0, shflxor64(min0, 16));
  min1 = minu64(min1, shflxor64(min1, 16));
#pragma unroll
  for (int m2 = 16; m2 >= 1; m2 >>= 1) condAcc |= __shfl_xor(condAcc, m2, 32);

  if (lane == 0) atomicOr(&anyPos[b], condAcc);
  if (lane < 16) {
    atomicMin(&minbuf[b * GG + lane], min0);        // g = 0..15
  } else if (lane < GG) {
    atomicMin(&minbuf[b * GG + lane], min1);        // g = 16..23 (column lane-16)
  }
}

__global__ void __launch_bounds__(256)
lf_passB(const float* __restrict__ cls, const float* __restrict__ params,
         const float* __restrict__ pinit, const float* __restrict__ tgtp,
         const float* __restrict__ pts, const unsigned long long* __restrict__ minbuf,
         const unsigned* __restrict__ anyPos, double* __restrict__ partial) {
  const int b     = blockIdx.x / CHUNKS;
  const int chunk = blockIdx.x % CHUNKS;
  __shared__ float sT[GG], sR[GG];
  __shared__ unsigned sAmin[GG];
  __shared__ unsigned char sValid[GG];
  __shared__ unsigned sAny;
  const int t = threadIdx.x;
  if (t < GG) {
    float th = tgtp[(b * GG + t) * 2 + 0];
    float ra = tgtp[(b * GG + t) * 2 + 1];
    sT[t] = (th + 90.0f) * (1.0f / 180.0f);
    sR[t] = (ra + 400.0f) * (1.0f / 800.0f);
    sAmin[t]  = (unsigned)(minbuf[b * GG + t] & 0xffffffffull);
    sValid[t] = (pts[((size_t)b * GG + t) * 4] != PADV) ? 1 : 0;
  }
  if (t == GG) sAny = anyPos[b];
  __syncthreads();

  float tT[GG], tR[GG]; unsigned am[GG];
  unsigned validM = 0u;
#pragma unroll
  for (int g = 0; g < GG; ++g) {
    tT[g] = sT[g]; tR[g] = sR[g]; am[g] = sAmin[g];
    validM |= ((unsigned)sValid[g]) << g;
  }
  const unsigned anyM = sAny;

  const float2* cls2 = (const float2*)cls;
  const float2* par2 = (const float2*)params;
  const float2* pi2  = (const float2*)pinit;

  double accReg = 0.0, accCls = 0.0;
  for (int k = 0; k < 4; ++k) {
    const unsigned n  = (unsigned)(chunk * ROWS_PER_BLOCK + k * 256 + t);
    const size_t  idx = (size_t)b * NN + n;
    float2 pi = pi2[idx];
    float2 pp = par2[idx];
    float2 cv = cls2[idx];
    unsigned posM = 0u;
    float reg = 0.0f;
#pragma unroll
    for (int g = 0; g < GG; ++g) {
      bool c   = (fabsf(tT[g] - pi.x) < THT) && (fabsf(tR[g] - pi.y) < THR);
      bool fb  = (n == am[g]);
      bool ap  = (anyM >> g) & 1u;
      bool v   = (validM >> g) & 1u;
      bool pos = v && (ap ? c : fb);
      posM |= ((unsigned)pos) << g;
      if (pos) {
        float dt = tT[g] - pp.x;
        float dr = tR[g] - pp.y;
        reg += 0.5f * (dt * dt + dr * dr);   // mean over 2 params
      }
    }
    accReg += (double)reg;
    // 2-class softmax focal loss (gamma=2) at class gt = (posM != 0)
    const float x0 = cv.x, x1 = cv.y;
    const float mx  = fmaxf(x0, x1);
    const float lse = mx + logf(expf(x0 - mx) + expf(x1 - mx));
    const float lp  = (posM ? x1 : x0) - lse;
    const float p   = expf(lp);
    const float om  = 1.0f - p;
    accCls += (double)(om * om * lp);
  }

  __shared__ double redR[256], redC[256];
  redR[t] = accReg; redC[t] = accCls;
  __syncthreads();
  for (int s = 128; s > 0; s >>= 1) {
    if (t < s) { redR[t] += redR[t + s]; redC[t] += redC[t + s]; }
    __syncthreads();
  }
  if (t == 0) { partial[2 * blockIdx.x] = redR[0]; partial[2 * blockIdx.x + 1] = redC[0]; }
}

__global__ void __launch_bounds__(256)
lf_final(const double* __restrict__ partial, int nblk, float* __restrict__ out) {
  __shared__ double redR[256], redC[256];
  const int t = threadIdx.x;
  double r = 0.0, c = 0.0;
  for (int i = t; i < nblk; i += 256) { r += partial[2 * i]; c += partial[2 * i + 1]; }
  redR[t] = r; redC[t] = c;
  __syncthreads();
  for (int s = 128; s > 0; s >>= 1) {
    if (t < s) { redR[t] += redR[t + s]; redC[t] += redC[t + s]; }
    __syncthreads();
  }
  if (t == 0) {
    double lossCls = -(redC[0] / ((double)BB * (double)NN));
    double lossReg = redR[0] / (double)BB;
    out[0] = (float)(2.0 * lossCls);   // W_CLS * loss_cls
    out[1] = (float)(5.0 * lossReg);   // W_REG * loss_reg
  }
}

extern "C" void kernel_launch(void* const* d_in, const int* in_sizes, int n_in,
                              void* d_out, int out_size, void* d_ws, size_t ws_size,
                              hipStream_t stream) {
  (void)in_sizes; (void)n_in; (void)out_size; (void)ws_size;
  const float* cls    = (const float*)d_in[0];
  const float* params = (const float*)d_in[1];
  const float* pinit  = (const float*)d_in[2];
  const float* tgtp   = (const float*)d_in[3];
  const float* pts    = (const float*)d_in[4];
  float* out = (float*)d_out;

  // Workspace layout (init'd every call; harness poisons with 0xAA):
  //   [0)     : u64 minbuf[B*G]   (12288 B)
  //   [12288) : u32 anyPos[B]     (256 B)
  //   [12544) : double partial[B*CHUNKS*2] (16384 B)
  unsigned long long* minbuf = (unsigned long long*)d_ws;
  unsigned* anyPos = (unsigned*)((char*)d_ws + (size_t)BB * GG * sizeof(unsigned long long));
  double* partial  = (double*)((char*)d_ws + (size_t)BB * GG * sizeof(unsigned long long)
                                           + (size_t)BB * sizeof(unsigned));

  lf_init<<<(BB * GG + 255) / 256, 256, 0, stream>>>(minbuf, anyPos);
  lf_passA<<<BB * CHUNKS, 256, 0, stream>>>(pinit, tgtp, minbuf, anyPos);
  lf_passB<<<BB * CHUNKS, 256, 0, stream>>>(cls, params, pinit, tgtp, pts, minbuf, anyPos, partial);
  lf_final<<<1, 256, 0, stream>>>(partial, BB * CHUNKS, out);
}